// Hetro_GIN_7541962572002
// MI455X (gfx1250) — compile-verified
//
#include <hip/hip_runtime.h>
#include <hip/hip_bf16.h>

typedef float v2f __attribute__((ext_vector_type(2)));
typedef float v8f __attribute__((ext_vector_type(8)));

#define NPATH 100000
#define NLINK 150000
#define NNODE 50000

#if defined(__HIP_DEVICE_COMPILE__) && __has_builtin(__builtin_amdgcn_global_load_async_to_lds_b128)
#define HAVE_ASYNC_LDS 1
#else
#define HAVE_ASYNC_LDS 0
#endif

#if HAVE_ASYNC_LDS
// builtin param 0 prints as: "__attribute__((__vector_size__(4 * sizeof(int)))) int __device__ *"
// i.e. non-const int4 (gcc vector_size) pointer in AS(1); LDS side is AS(3).
typedef int v4i_ls __attribute__((vector_size(16)));
typedef v4i_ls __attribute__((address_space(1)))* async_gptr;
typedef v4i_ls __attribute__((address_space(3)))* async_lptr;
#endif

__device__ __forceinline__ void wait_async_then_barrier() {
#if defined(__HIP_DEVICE_COMPILE__)
#if __has_builtin(__builtin_amdgcn_s_wait_asynccnt)
  __builtin_amdgcn_s_wait_asynccnt(0);
#elif HAVE_ASYNC_LDS
  asm volatile("s_wait_asynccnt 0" ::: "memory");
#endif
#endif
  __syncthreads();
}

// Stage nflt floats of W (contiguous) into LDS, 16B per lane per step.
__device__ __forceinline__ void stage_w(const float* __restrict__ W, float* sW, int nflt) {
  for (int idx = threadIdx.x * 4; idx < nflt; idx += blockDim.x * 4) {
#if HAVE_ASYNC_LDS
    __builtin_amdgcn_global_load_async_to_lds_b128(
        (async_gptr)(W + idx), (async_lptr)(sW + idx), 0, 0);
#else
    float4 v = *(const float4*)(W + idx);
    *(float4*)(sW + idx) = v;
#endif
  }
  wait_async_then_barrier();
}

// ---------------------------------------------------------------- utility
__global__ void k_zero(float* __restrict__ p, int n) {
  int i = blockIdx.x * blockDim.x + threadIdx.x;
  if (i < n) p[i] = 0.0f;
}

// X[row, col] = (col < dagg) ? 0 : (1+eps)*xdst[row, col-dagg]
__global__ void k_prep(float* __restrict__ X, const float* __restrict__ xdst,
                       const float* __restrict__ epsp,
                       long long total, int fin, int dagg, int dself) {
  long long i = (long long)blockIdx.x * blockDim.x + threadIdx.x;
  if (i >= total) return;
  int col = (int)(i % fin);
  long long row = i / fin;
  float v = 0.0f;
  if (col >= dagg) v = (1.0f + epsp[0]) * xdst[row * dself + (col - dagg)];
  X[i] = v;
}

// scatter-add, 4 floats per thread: X[dst[e], f..f+3] += xsrc[src[e], f..f+3]
__global__ void k_scatter(float* __restrict__ X, int ldX,
                          const float* __restrict__ xsrc, int dsrc,
                          const int* __restrict__ src, const int* __restrict__ dst,
                          long long total4) {
  long long i = (long long)blockIdx.x * blockDim.x + threadIdx.x;
  if (i >= total4) return;
  int q = dsrc >> 2;
  int f = (int)(i % q) * 4;
  long long e = i / q;
  const float* s = xsrc + (size_t)src[e] * dsrc + f;
  float* d = X + (size_t)dst[e] * ldX + f;
  float4 v = *(const float4*)s;
  atomicAdd(d + 0, v.x);
  atomicAdd(d + 1, v.y);
  atomicAdd(d + 2, v.z);
  atomicAdd(d + 3, v.w);
}

// ---------------------------------------------------------------- GEMM1 (+ BN stats)
// H[N, NT*16] = X[N,K] @ W[K, NT*16] + bias ; accumulate column sum/sumsq.
// 256 threads = 8 waves; W staged once per block in LDS; each wave owns a 16-row tile.
template <int NT>
__global__ void k_gemm1(const float* __restrict__ X, int K,
                        const float* __restrict__ W,
                        const float* __restrict__ bias,
                        float* __restrict__ H, int N,
                        float* __restrict__ s1, float* __restrict__ s2) {
  extern __shared__ float sW[];
  const int ldW = NT * 16;
  stage_w(W, sW, K * ldW);

  int wave = threadIdx.x >> 5;
  int lane = threadIdx.x & 31;
  int half = lane >> 4;
  int l15 = lane & 15;
  int m0 = (blockIdx.x * 8 + wave) * 16;
  int arow = m0 + l15;
  const bool av = arow < N;
  const float* Xr = X + (size_t)arow * K;

  v8f acc[NT];
#pragma unroll
  for (int t = 0; t < NT; ++t)
#pragma unroll
    for (int r = 0; r < 8; ++r) acc[t][r] = 0.0f;

  for (int k0 = 0; k0 < K; k0 += 4) {
    int ka = k0 + 2 * half;
    v2f a;
    a.x = av ? Xr[ka] : 0.0f;
    a.y = av ? Xr[ka + 1] : 0.0f;
    if (av && k0 + 16 < K) __builtin_prefetch(Xr + ka + 16, 0, 0);
#pragma unroll
    for (int t = 0; t < NT; ++t) {
      int n = t * 16 + l15;
      v2f b;
      b.x = sW[ka * ldW + n];
      b.y = sW[(ka + 1) * ldW + n];
      acc[t] = __builtin_amdgcn_wmma_f32_16x16x4_f32(
          false, a, false, b, (short)0, acc[t], false, false);
    }
  }

#pragma unroll
  for (int t = 0; t < NT; ++t) {
    int n = t * 16 + l15;
    float bb = bias[n];
    float ls1 = 0.0f, ls2 = 0.0f;
#pragma unroll
    for (int r = 0; r < 8; ++r) {
      int m = m0 + 8 * half + r;
      if (m < N) {
        float d = acc[t][r] + bb;
        H[(size_t)m * ldW + n] = d;
        ls1 += d;
        ls2 += d * d;
      }
    }
    ls1 += __shfl_down(ls1, 16);
    ls2 += __shfl_down(ls2, 16);
    if (half == 0) {
      atomicAdd(&s1[n], ls1);
      atomicAdd(&s2[n], ls2);
    }
  }
}

// BN affine finalize: scale = g*rsqrt(var+eps), shift = b - mu*scale
__global__ void k_bnfin(const float* __restrict__ s1, const float* __restrict__ s2,
                        const float* __restrict__ g, const float* __restrict__ b,
                        float invN, float* __restrict__ sc, float* __restrict__ sh) {
  int n = threadIdx.x;
  float mu = s1[n] * invN;
  float var = fmaxf(s2[n] * invN - mu * mu, 0.0f);
  float inv = rsqrtf(var + 1e-5f);
  float s = g[n] * inv;
  sc[n] = s;
  sh[n] = b[n] - mu * s;
}

// ---------------------------------------------------------------- GEMM2
// OUT[N,128] (=/+=) relu(H*scale+shift)[N,128] @ W2[128,128] + b2 ; optional relu
__global__ void k_gemm2(const float* __restrict__ H,
                        const float* __restrict__ sc, const float* __restrict__ sh,
                        const float* __restrict__ W2, const float* __restrict__ b2,
                        float* __restrict__ OUT, int N, int accum, int dorelu) {
  extern __shared__ float sW[];
  stage_w(W2, sW, 128 * 128);

  int wave = threadIdx.x >> 5;
  int lane = threadIdx.x & 31;
  int half = lane >> 4;
  int l15 = lane & 15;
  int m0 = (blockIdx.x * 8 + wave) * 16;
  int arow = m0 + l15;
  const bool av = arow < N;
  const float* Hr = H + (size_t)arow * 128;

  v8f acc[8];
#pragma unroll
  for (int t = 0; t < 8; ++t)
#pragma unroll
    for (int r = 0; r < 8; ++r) acc[t][r] = 0.0f;

  for (int k0 = 0; k0 < 128; k0 += 4) {
    int ka = k0 + 2 * half;
    v2f a;
    if (av) {
      float t0 = Hr[ka] * sc[ka] + sh[ka];
      float t1 = Hr[ka + 1] * sc[ka + 1] + sh[ka + 1];
      a.x = fmaxf(t0, 0.0f);
      a.y = fmaxf(t1, 0.0f);
    } else {
      a.x = 0.0f;
      a.y = 0.0f;
    }
#pragma unroll
    for (int t = 0; t < 8; ++t) {
      int n = t * 16 + l15;
      v2f b;
      b.x = sW[ka * 128 + n];
      b.y = sW[(ka + 1) * 128 + n];
      acc[t] = __builtin_amdgcn_wmma_f32_16x16x4_f32(
          false, a, false, b, (short)0, acc[t], false, false);
    }
  }

#pragma unroll
  for (int t = 0; t < 8; ++t) {
    int n = t * 16 + l15;
    float bb = b2[n];
#pragma unroll
    for (int r = 0; r < 8; ++r) {
      int m = m0 + 8 * half + r;
      if (m < N) {
        size_t idx = (size_t)m * 128 + n;
        float d = acc[t][r] + bb;
        if (accum) d += OUT[idx];
        if (dorelu) d = fmaxf(d, 0.0f);
        OUT[idx] = d;
      }
    }
  }
}

// ---------------------------------------------------------------- post head
// out[i] = relu( sum_j prelu(bn(PH[i,j])) * W2[j] + b2 )
__global__ void k_post(const float* __restrict__ PH, const float* __restrict__ sc,
                       const float* __restrict__ sh, const float* __restrict__ ap,
                       const float* __restrict__ W2, const float* __restrict__ b2,
                       float* __restrict__ out, int N) {
  int i = blockIdx.x * blockDim.x + threadIdx.x;
  if (i >= N) return;
  float a = ap[0];
  float acc = b2[0];
  const float* h = PH + (size_t)i * 16;
#pragma unroll
  for (int j = 0; j < 16; ++j) {
    float t = h[j] * sc[j] + sh[j];
    t = (t >= 0.0f) ? t : a * t;
    acc += t * W2[j];
  }
  out[i] = fmaxf(acc, 0.0f);
}

// ---------------------------------------------------------------- host
struct GinP {
  const float *bnb, *bng, *eps, *W1, *b1, *W2, *b2;
};

extern "C" void kernel_launch(void* const* d_in, const int* in_sizes, int n_in,
                              void* d_out, int out_size, void* d_ws, size_t ws_size,
                              hipStream_t stream) {
  const float* x_path = (const float*)d_in[0];
  const float* x_link = (const float*)d_in[1];
  const float* x_node = (const float*)d_in[2];
  const int* pul_src = (const int*)d_in[3];
  const int* pul_dst = (const int*)d_in[4];
  const int* lip_src = (const int*)d_in[5];
  const int* lip_dst = (const int*)d_in[6];
  const int* lcn_src = (const int*)d_in[7];
  const int* lcn_dst = (const int*)d_in[8];
  const int* nhl_src = (const int*)d_in[9];
  const int* nhl_dst = (const int*)d_in[10];
  const long long E = (long long)in_sizes[3];

  // ---- param leaves, jax pytree order:
  // layers[0]{lcn,lip,nhl,pul} layers[1]{lcn,lip,nhl,pul} each {bn.b,bn.g,eps,lin1.W,lin1.b,lin2.W,lin2.b}
  // post {a, bn.b, bn.g, lin1.W, lin1.b, lin2.W, lin2.b}
  const float* leaf[63];
  if (n_in >= 74) {
    for (int i = 0; i < 63; ++i) leaf[i] = (const float*)d_in[11 + i];
  } else {
    // single flat params buffer at d_in[11]
    static const int fins[8] = {80, 96, 80, 96, 128, 128, 128, 128};
    const float* base = (const float*)d_in[11];
    size_t off = 0;
    int li = 0;
    for (int g = 0; g < 8; ++g) {
      size_t sz[7] = {128, 128, 1, (size_t)fins[g] * 128, 128, 128 * 128, 128};
      for (int j = 0; j < 7; ++j) { leaf[li++] = base + off; off += sz[j]; }
    }
    size_t psz[7] = {1, 16, 16, 128 * 16, 16, 16, 1};
    for (int j = 0; j < 7; ++j) { leaf[li++] = base + off; off += psz[j]; }
  }
  auto G = [&](int gi) {
    const float* const* L = leaf + gi * 7;
    GinP g{L[0], L[1], L[2], L[3], L[4], L[5], L[6]};
    return g;
  };
  // gin order: 0=lcn0 1=lip0 2=nhl0 3=pul0 4=lcn1 5=lip1 6=nhl1 7=pul1

  // ---- workspace layout (floats)
  float* ws = (float*)d_ws;
  size_t off = 0;
  float* EMB_P0 = ws + off; off += (size_t)NPATH * 128;
  float* EMB_L0 = ws + off; off += (size_t)NLINK * 128;
  float* EMB_N0 = ws + off; off += (size_t)NNODE * 128;
  float* EMB_L1 = ws + off; off += (size_t)NLINK * 128;
  float* XB     = ws + off; off += (size_t)NLINK * 128;  // gin input scratch (also post PH)
  float* HB     = ws + off; off += (size_t)NLINK * 128;  // gin hidden scratch
  float* S1 = ws + off; off += 128;
  float* S2 = ws + off; off += 128;
  float* SC = ws + off; off += 128;
  float* SH = ws + off; off += 128;
  (void)ws_size; (void)out_size;

  auto run_gin = [&](GinP g, const float* xsrc, int dsrc, const float* xdst, int dself,
                     int dagg, int fin, const int* src, const int* dst, int Ndst,
                     float* outb, int accum, int dorelu) {
    long long tp = (long long)Ndst * fin;
    k_prep<<<(unsigned)((tp + 255) / 256), 256, 0, stream>>>(XB, xdst, g.eps, tp, fin, dagg, dself);
    long long ts = E * (dsrc / 4);
    k_scatter<<<(unsigned)((ts + 255) / 256), 256, 0, stream>>>(XB, fin, xsrc, dsrc, src, dst, ts);
    k_zero<<<1, 256, 0, stream>>>(S1, 256);  // S1 and S2 contiguous
    k_gemm1<8><<<(Ndst + 127) / 128, 256, (size_t)fin * 128 * 4, stream>>>(
        XB, fin, g.W1, g.b1, HB, Ndst, S1, S2);
    k_bnfin<<<1, 128, 0, stream>>>(S1, S2, g.bng, g.bnb, 1.0f / (float)Ndst, SC, SH);
    k_gemm2<<<(Ndst + 127) / 128, 256, (size_t)128 * 128 * 4, stream>>>(
        HB, SC, SH, g.W2, g.b2, outb, Ndst, accum, dorelu);
  };

  // ---- layer 0 (concat=True): dagg = dsrc
  run_gin(G(1), x_link, 64, x_path, 32, 64, 96, lip_src, lip_dst, NPATH, EMB_P0, 0, 1);  // lip -> path
  run_gin(G(0), x_link, 64, x_node, 16, 64, 80, lcn_src, lcn_dst, NNODE, EMB_N0, 0, 1);  // lcn -> node
  run_gin(G(3), x_path, 32, x_link, 64, 32, 96, pul_src, pul_dst, NLINK, EMB_L0, 0, 0);  // pul -> link
  run_gin(G(2), x_node, 16, x_link, 64, 16, 80, nhl_src, nhl_dst, NLINK, EMB_L0, 1, 1);  // nhl += , relu

  // ---- layer 1: output only depends on link embedding -> lip/lcn are dead code
  run_gin(G(7), EMB_P0, 128, EMB_L0, 128, 0, 128, pul_src, pul_dst, NLINK, EMB_L1, 0, 0);
  run_gin(G(6), EMB_N0, 128, EMB_L0, 128, 0, 128, nhl_src, nhl_dst, NLINK, EMB_L1, 1, 1);

  // ---- post head on links
  const float* p_a   = leaf[56];
  const float* p_bnb = leaf[57];
  const float* p_bng = leaf[58];
  const float* p_W1  = leaf[59];
  const float* p_b1  = leaf[60];
  const float* p_W2  = leaf[61];
  const float* p_b2  = leaf[62];
  float* PH = XB;  // reuse scratch as [NLINK,16]
  k_zero<<<1, 256, 0, stream>>>(S1, 256);
  k_gemm1<1><<<(NLINK + 127) / 128, 256, (size_t)128 * 16 * 4, stream>>>(
      EMB_L1, 128, p_W1, p_b1, PH, NLINK, S1, S2);
  k_bnfin<<<1, 16, 0, stream>>>(S1, S2, p_bng, p_bnb, 1.0f / (float)NLINK, SC, SH);
  k_post<<<(NLINK + 255) / 256, 256, 0, stream>>>(PH, SC, SH, p_a, p_W2, p_b2, (float*)d_out, NLINK);
}